// CustomDropout_13477607375212
// MI455X (gfx1250) — compile-verified
//
#include <hip/hip_runtime.h>
#include <hip/hip_bf16.h>
#include <stdint.h>

typedef float v2f __attribute__((ext_vector_type(2)));
typedef float v8f __attribute__((ext_vector_type(8)));

#define B_    32
#define H_    12
#define N_    577
#define C_    768
#define NN_   332929u                 // N*N
#define PLANE_ 3995148u               // H*NN (elements between consecutive b, same h)
#define CHUNKS_ 5202u                 // (NN-1)/64 full 64-elem chunks per plane
#define SIDE_F 24.020824298928627f    // sqrt(577)
#define HALF_U 7090176u               // H*B*N*64 / 2
#define DIST_OFF 256                  // float offset of dist table in ws
#define BPH   128                     // blocks per head for the reduction

// ws float layout: [0,12) num | [12,24) den | [24,36) p | [36,48) scale | [256,256+NN) dist
// ws bytes needed: ~1.34 MB

// ---------------- kernel 1: init accumulators + dist table ----------------
__global__ void k_init(float* __restrict__ ws) {
    uint32_t t = blockIdx.x * 256u + threadIdx.x;
    if (t < 48u) ws[t] = 0.0f;
    if (t < NN_) {
        uint32_t i = t / N_;
        uint32_t j = t - i * N_;
        float d  = (float)j - (float)i;
        float fd = floorf(d / SIDE_F);        // floor_divide(d, side)
        float m  = d - fd * SIDE_F;           // d % side (sign of divisor)
        ws[DIST_OFF + t] = sqrtf(m * m + fd * fd);
    }
}

// ---------------- kernel 2: WMMA-based weighted reduction ----------------
// num[h] = sum attn*dist, den[h] = sum attn over (b,i,j).
// B-operand = ones, so each WMMA folds 64 fp32 values into the accumulator
// (every row-sum replicated across 16 columns -> * 1/16 at the end).
// Loop order: o-chunk outer (dist loaded once), b inner at constant stride ->
// no division, no guards, EXEC all-ones throughout.
__global__ void k_reduce(const float* __restrict__ attn, float* __restrict__ ws) {
    const float* __restrict__ dist = ws + DIST_OFF;
    const uint32_t h    = blockIdx.x / BPH;
    const uint32_t blk  = blockIdx.x % BPH;
    const uint32_t lane = threadIdx.x & 31u;
    const uint32_t wav  = threadIdx.x >> 5;          // 0..7
    const uint32_t w    = blk * 8u + wav;            // wave id within head, 0..1023
    const float* __restrict__ attn_h = attn + (size_t)h * NN_;

    v2f ones  = {1.0f, 1.0f};
    v8f accD0 = {0.f,0.f,0.f,0.f,0.f,0.f,0.f,0.f};
    v8f accD1 = accD0, accN0 = accD0, accN1 = accD0;

    for (uint32_t c = w; c < CHUNKS_; c += BPH * 8u) {
        uint32_t o0 = c * 64u + lane;                // coalesced 128B stream
        uint32_t o1 = o0 + 32u;                      // second 128B stream
        float d0 = dist[o0];
        float d1 = dist[o1];
        uint32_t i0 = o0, i1 = o1;                   // 32-bit voffsets (max ~124M elems)
        #pragma unroll 4
        for (int b = 0; b < B_; b += 2) {
            float a0 = __builtin_nontemporal_load(attn_h + i0);
            float a1 = __builtin_nontemporal_load(attn_h + i1);
            float a2 = __builtin_nontemporal_load(attn_h + i0 + PLANE_);
            float a3 = __builtin_nontemporal_load(attn_h + i1 + PLANE_);
            v2f A0 = {a0, a1}, P0 = {a0 * d0, a1 * d1};
            v2f A1 = {a2, a3}, P1 = {a2 * d0, a3 * d1};
            accD0 = __builtin_amdgcn_wmma_f32_16x16x4_f32(false, A0, false, ones,
                                                          (short)0, accD0, false, false);
            accN0 = __builtin_amdgcn_wmma_f32_16x16x4_f32(false, P0, false, ones,
                                                          (short)0, accN0, false, false);
            accD1 = __builtin_amdgcn_wmma_f32_16x16x4_f32(false, A1, false, ones,
                                                          (short)0, accD1, false, false);
            accN1 = __builtin_amdgcn_wmma_f32_16x16x4_f32(false, P1, false, ones,
                                                          (short)0, accN1, false, false);
            i0 += 2u * PLANE_;
            i1 += 2u * PLANE_;
        }
    }

    float sN = 0.f, sD = 0.f;
    #pragma unroll
    for (int i = 0; i < 8; ++i) {
        sN += accN0[i] + accN1[i];
        sD += accD0[i] + accD1[i];
    }
    for (int off = 16; off > 0; off >>= 1) {
        sN += __shfl_xor(sN, off, 32);
        sD += __shfl_xor(sD, off, 32);
    }
    if (lane == 0) {
        atomicAdd(ws +       h, sN * 0.0625f);   // /16: row sums replicated over 16 cols
        atomicAdd(ws + 12u + h, sD * 0.0625f);
    }
}

// ---------------- kernel 3: plane-tail fixup + per-head probability ----------------
// The last element of each plane (o = NN-1, i=j=576 -> dist = 0) was skipped by
// k_reduce; it contributes only to den.
__global__ void k_proba(const float* __restrict__ attn, float* __restrict__ ws) {
    int t = threadIdx.x;
    if (t < H_) {
        float den = ws[12 + t];
        for (int b = 0; b < B_; ++b)
            den += attn[(size_t)(b * H_ + t) * NN_ + (NN_ - 1u)];
        float avg = ws[t] / den;
        float p   = 0.5f * expf(-5.0f * avg / SIDE_F);
        ws[24 + t] = p;
        ws[36 + t] = 1.0f / (1.0f - p);
    }
}

// ---------------- kernel 4: threefry dropout apply ----------------
// One block per (b,n) row, 384 threads: thread c covers output columns c and
// c+384, which share one threefry2x32 block (counter i and i+HALF map to heads
// h and h+6 at the same (b,n,ch)).
__device__ __forceinline__ uint32_t rotl32(uint32_t v, int r) {
    return (v << r) | (v >> (32 - r));
}

__global__ void k_dropout(const float* __restrict__ x, const float* __restrict__ ws,
                          float* __restrict__ out) {
    uint32_t row = blockIdx.x;            // b*N + n   (scalar)
    uint32_t c   = threadIdx.x;           // 0..383
    uint32_t n   = row % N_;
    uint32_t b   = row / N_;
    uint32_t h   = c >> 6;                // 0..5
    uint32_t ch  = c & 63u;
    uint32_t i   = (((h * B_ + b) * N_ + n) << 6) + ch;   // u-index < HALF_U

    // threefry2x32-20, key = (0, 42)  [jax.random.key(42)]
    const uint32_t k0 = 0u, k1 = 42u, k2 = 0u ^ 42u ^ 0x1BD11BDAu;
    uint32_t x0 = i + k0, x1 = (i + HALF_U) + k1;
#define TF_R(r) { x0 += x1; x1 = rotl32(x1, r); x1 ^= x0; }
    TF_R(13) TF_R(15) TF_R(26) TF_R(6)   x0 += k1; x1 += k2 + 1u;
    TF_R(17) TF_R(29) TF_R(16) TF_R(24)  x0 += k2; x1 += k0 + 2u;
    TF_R(13) TF_R(15) TF_R(26) TF_R(6)   x0 += k0; x1 += k1 + 3u;
    TF_R(17) TF_R(29) TF_R(16) TF_R(24)  x0 += k1; x1 += k2 + 4u;
    TF_R(13) TF_R(15) TF_R(26) TF_R(6)   x0 += k2; x1 += k0 + 5u;
#undef TF_R

    float u0 = __uint_as_float((x0 >> 9) | 0x3f800000u) - 1.0f;
    float u1 = __uint_as_float((x1 >> 9) | 0x3f800000u) - 1.0f;

    uint32_t base = row * C_ + c;         // out/x flat index (same position!)
    float p0 = ws[24 + h],     s0 = ws[36 + h];
    float p1 = ws[24 + h + 6], s1 = ws[36 + h + 6];
    float v0 = (u0 >= p0) ? x[base]        * s0 : 0.0f;
    float v1 = (u1 >= p1) ? x[base + 384u] * s1 : 0.0f;
    __builtin_nontemporal_store(v0, out + base);
    __builtin_nontemporal_store(v1, out + base + 384u);
}

// ---------------- launch ----------------
extern "C" void kernel_launch(void* const* d_in, const int* in_sizes, int n_in,
                              void* d_out, int out_size, void* d_ws, size_t ws_size,
                              hipStream_t stream) {
    const float* x    = (const float*)d_in[0];
    const float* attn = (const float*)d_in[1];
    float* ws  = (float*)d_ws;
    float* out = (float*)d_out;

    k_init   <<<(NN_ + 255u) / 256u, 256, 0, stream>>>(ws);
    k_reduce <<<H_ * BPH,            256, 0, stream>>>(attn, ws);
    k_proba  <<<1,                    32, 0, stream>>>(attn, ws);
    k_dropout<<<B_ * N_,             384, 0, stream>>>(x, ws, out);  // 18464 * 384 * 2 = B*N*C
}